// SolventAccessibility_54803782697319
// MI455X (gfx1250) — compile-verified
//
#include <hip/hip_runtime.h>
#include <stdint.h>

// ---- problem constants (match reference) -----------------------------------
#define PADV    (-999)
#define BATCHN  8
#define NCHAINN 2
#define SEQLEN  1024
#define NALTN   3
#define NRESN   20
#define GLYV    7
#define NATOMSN 2000000
#define NBINSN  (BATCHN * NCHAINN * SEQLEN * NALTN)   // 49152

// ---- CDNA5 async-to-LDS availability (device pass only) --------------------
#if defined(__has_builtin)
#if __has_builtin(__builtin_amdgcn_global_load_async_to_lds_b32)
#define HAS_ASYNC_LDS 1
#endif
#endif

#if defined(HAS_ASYNC_LDS)
// Parameter types per hipcc diagnostic: global-AS int* (non-const), LDS-AS int*.
typedef __attribute__((address_space(1))) int* as1_int_ptr;
typedef __attribute__((address_space(3))) int* as3_int_ptr;

__device__ __forceinline__ void async_cp_b32(const void* g, const int* l) {
    __builtin_amdgcn_global_load_async_to_lds_b32(
        (as1_int_ptr)(uintptr_t)g,
        (as3_int_ptr)(uint32_t)(uintptr_t)l, 0, 0);
}
#endif

__device__ __forceinline__ void wait_async_le8() {
#if defined(__has_builtin)
#if __has_builtin(__builtin_amdgcn_s_wait_asynccnt)
    __builtin_amdgcn_s_wait_asynccnt(8);
#else
    asm volatile("s_wait_asynccnt 8" ::: "memory");
#endif
#else
    asm volatile("s_wait_asynccnt 8" ::: "memory");
#endif
}
__device__ __forceinline__ void wait_async_le0() {
#if defined(__has_builtin)
#if __has_builtin(__builtin_amdgcn_s_wait_asynccnt)
    __builtin_amdgcn_s_wait_asynccnt(0);
#else
    asm volatile("s_wait_asynccnt 0" ::: "memory");
#endif
#else
    asm volatile("s_wait_asynccnt 0" ::: "memory");
#endif
}

// ---- kernel A: init outputs and seq scratch (exactly NBINS threads) --------
__global__ void __launch_bounds__(256)
sa_init_kernel(float* __restrict__ out, int* __restrict__ seq) {
    const int t = blockIdx.x * blockDim.x + threadIdx.x;   // 0 .. NBINS-1
    out[t]          = 0.0f;
    out[NBINSN + t] = 0.0f;
    seq[t]          = PADV;
}

// ---- per-atom scatter body --------------------------------------------------
__device__ __forceinline__ void process_atom(
    int atname, int resnum, int chain, int batch, int resname,
    float c0, float c1, float c2,
    uint8_t a0, uint8_t a1, uint8_t a2,
    float* __restrict__ outMC, float* __restrict__ outSC, int* __restrict__ seq)
{
    const bool valid = (atname != PADV);
    const bool bb    = ((unsigned)atname) <= 3u;            // BACKBONE = {0,1,2,3}
    const int  base  = (batch * NCHAINN + chain) * SEQLEN + resnum;
    float* __restrict__ dst = bb ? outMC : outSC;

    if (valid) {
        if (a0) { atomicAdd(dst + base * 3 + 0, c0); if (bb) seq[base * 3 + 0] = resname; }
        if (a1) { atomicAdd(dst + base * 3 + 1, c1); if (bb) seq[base * 3 + 1] = resname; }
        if (a2) { atomicAdd(dst + base * 3 + 2, c2); if (bb) seq[base * 3 + 2] = resname; }
    }
}

// ---- kernel B: scatter-add with double-buffered async-LDS staging ----------
__global__ void __launch_bounds__(256)
sa_scatter_kernel(const float*   __restrict__ contRat,  // [NATOMS,3] f32
                  const int*     __restrict__ adesc,    // [NATOMS,5] int32
                  const uint8_t* __restrict__ alts,     // [NATOMS,3] bool(u8)
                  float*         __restrict__ out,      // [2*NBINS]
                  int*           __restrict__ seq)      // [NBINS]
{
    float* outMC = out;
    float* outSC = out + NBINSN;

#if defined(HAS_ASYNC_LDS)
    // Per wave32: two 256-dword slabs (desc 5 dw/atom at [0..159],
    // contRat 3 dw/atom at [160..255]). 8 waves * 2 bufs * 1 KB = 16 KB/block.
    // Stage with 8 coalesced GLOBAL_LOAD_ASYNC_TO_LDS_B32 per lane per chunk,
    // double-buffered: ASYNCcnt completes in order, so after issuing the next
    // chunk's 8 copies, s_wait_asynccnt 8 guarantees the current chunk landed.
    __shared__ int lds[8 * 2 * 256];
    const int lane    = threadIdx.x & 31;
    const int wv      = threadIdx.x >> 5;
    const int wid     = blockIdx.x * (blockDim.x >> 5) + wv;
    const int nwaves  = gridDim.x * (blockDim.x >> 5);
    const int nchunks = NATOMSN / 32;                 // 62500, exact
    int* slab = lds + wv * 512;

    auto issue = [&](int chunk, int buf) {
        const int  base_atom = chunk * 32;
        const int*   gd = adesc   + (size_t)base_atom * 5;
        const float* gc = contRat + (size_t)base_atom * 3;
        int* db = slab + buf * 256;
#pragma unroll
        for (int j = 0; j < 5; ++j)
            async_cp_b32(gd + lane + 32 * j, db + lane + 32 * j);
#pragma unroll
        for (int j = 0; j < 3; ++j)
            async_cp_b32(gc + lane + 32 * j, db + 160 + lane + 32 * j);
    };

    int buf = 0;
    if (wid < nchunks) issue(wid, 0);
    for (int c = wid; c < nchunks; c += nwaves) {
        const int cnext = c + nwaves;
        if (cnext < nchunks) { issue(cnext, buf ^ 1); wait_async_le8(); }
        else                 { wait_async_le0(); }

        const int*   db = slab + buf * 256;
        const float* cb = (const float*)(db + 160);
        const int i = c * 32 + lane;

        process_atom(db[lane * 5 + 0], db[lane * 5 + 1], db[lane * 5 + 2],
                     db[lane * 5 + 3], db[lane * 5 + 4],
                     cb[lane * 3 + 0], cb[lane * 3 + 1], cb[lane * 3 + 2],
                     alts[i * 3 + 0], alts[i * 3 + 1], alts[i * 3 + 2],
                     outMC, outSC, seq);
        buf ^= 1;
    }
#else
    // Fallback: direct strided loads.
    const int tid    = blockIdx.x * blockDim.x + threadIdx.x;
    const int stride = gridDim.x * blockDim.x;
    for (int i = tid; i < NATOMSN; i += stride) {
        const int* r = adesc + (size_t)i * 5;
        process_atom(r[0], r[1], r[2], r[3], r[4],
                     contRat[i * 3 + 0], contRat[i * 3 + 1], contRat[i * 3 + 2],
                     alts[i * 3 + 0], alts[i * 3 + 1], alts[i * 3 + 2],
                     outMC, outSC, seq);
    }
#endif
}

// ---- kernel C: normalization + clip -----------------------------------------
__global__ void __launch_bounds__(256)
sa_norm_kernel(const float* __restrict__ mcmax, const float* __restrict__ mcmin,
               const float* __restrict__ scmax, const float* __restrict__ scmin,
               const int*   __restrict__ seq,   float* __restrict__ out)
{
    const int b = blockIdx.x * blockDim.x + threadIdx.x;   // 0 .. NBINS-1
    const int  s   = seq[b];
    const int  idx = min(max(s, 0), NRESN - 1);
    const bool has = (s != PADV);

    float mc = out[b];
    float sc = out[NBINSN + b];

    if (has) {
        mc = (mc - mcmin[idx]) / (mcmax[idx] - mcmin[idx]);
        sc = (s != GLYV) ? (sc - scmin[idx]) / (scmax[idx] - scmin[idx]) : 0.0f;
    }
    out[b]          = fminf(fmaxf(mc, 0.0f), 1.0f);
    out[NBINSN + b] = fminf(fmaxf(sc, 0.0f), 1.0f);
}

// ---- entry point -------------------------------------------------------------
extern "C" void kernel_launch(void* const* d_in, const int* in_sizes, int n_in,
                              void* d_out, int out_size, void* d_ws, size_t ws_size,
                              hipStream_t stream) {
    // setup_inputs() order:
    //   0: contRat f32[NATOMS,3]  1-4: res{MC,SC}_{max,min} f32[20]
    //   5: atom_description i32[NATOMS,5]  6: alternatives bool(u8)[NATOMS,3]
    const float*   contRat = (const float*)d_in[0];
    const float*   mcmax   = (const float*)d_in[1];
    const float*   mcmin   = (const float*)d_in[2];
    const float*   scmax   = (const float*)d_in[3];
    const float*   scmin   = (const float*)d_in[4];
    const int*     adesc   = (const int*)d_in[5];
    const uint8_t* alts    = (const uint8_t*)d_in[6];

    float* out = (float*)d_out;          // [rsaMC | rsaSC], 2*NBINS floats
    int*   seq = (int*)d_ws;             // NBINS ints of scratch

    sa_init_kernel<<<NBINSN / 256, 256, 0, stream>>>(out, seq);
    sa_scatter_kernel<<<2048, 256, 0, stream>>>(contRat, adesc, alts, out, seq);
    sa_norm_kernel<<<NBINSN / 256, 256, 0, stream>>>(mcmax, mcmin, scmax, scmin, seq, out);
}